// DCMLayer_19542101196947
// MI455X (gfx1250) — compile-verified
//
#include <hip/hip_runtime.h>

typedef __attribute__((ext_vector_type(2))) float v2f;
typedef __attribute__((ext_vector_type(8))) float v8f;
typedef __attribute__((ext_vector_type(4))) unsigned int u32x4;
typedef __attribute__((ext_vector_type(4))) int i32x4;
typedef __attribute__((ext_vector_type(8))) int i32x8;

#define N_  32
#define C_  256
#define H_  96
#define W_  96
#define HW_ (H_ * W_)
#define CM_ 64
#define P_  256
#define LDS_STRIDE 40   // 32 data dwords + 8 pad dwords (TDM pad) -> conflict-free

// tensor_load_to_lds arity differs between toolchains; guard on the TDM header.
#if __has_include(<hip/amd_detail/amd_gfx1250_TDM.h>)
#define TENSOR_LOAD_TO_LDS(g0, g1, g2, g3)                                     \
    __builtin_amdgcn_tensor_load_to_lds((g0), (g1), (g2), (g3),                \
                                        (i32x8){0, 0, 0, 0, 0, 0, 0, 0}, 0)
#else
#define TENSOR_LOAD_TO_LDS(g0, g1, g2, g3)                                     \
    __builtin_amdgcn_tensor_load_to_lds((g0), (g1), (g2), (g3), 0)
#endif

// ---------------------------------------------------------------------------
// Kernel 1a: adaptive 3x3 average pool of x -> xp stored transposed [n][3][3][c]
// ---------------------------------------------------------------------------
__global__ __launch_bounds__(288) void k_pool(const float* __restrict__ x,
                                              float* __restrict__ xp) {
    __shared__ float bins[9];
    const int nc  = blockIdx.x;          // n*C + c
    const int tid = threadIdx.x;         // 0..287
    if (tid < 9) bins[tid] = 0.f;
    __syncthreads();

    const int hh   = tid / 3;
    const int wseg = tid - hh * 3;
    const int bin  = (hh >> 5) * 3 + wseg;

    const float4* seg =
        (const float4*)(x + (size_t)nc * HW_ + hh * W_ + wseg * 32);
    float s = 0.f;
    #pragma unroll
    for (int i = 0; i < 8; ++i) {
        float4 v = seg[i];
        s += v.x + v.y + v.z + v.w;
    }
    atomicAdd(&bins[bin], s);
    __syncthreads();

    if (tid < 9) {
        int n = nc / C_, c = nc - n * C_;
        xp[((size_t)n * 9 + tid) * C_ + c] = bins[tid] * (1.f / 1024.f);
    }
}

// ---------------------------------------------------------------------------
// Kernel 1b: g[(n*64+o)*9 + bin] = dot(xp_t[n,bin,:], conv_w[o,:]) + conv_b[o]
// ---------------------------------------------------------------------------
__global__ __launch_bounds__(256) void k_gfilt(const float* __restrict__ xp,
                                               const float* __restrict__ w,
                                               const float* __restrict__ b,
                                               float* __restrict__ g) {
    const int out  = blockIdx.x * 8 + (threadIdx.x >> 5); // (n*64+o)*9+bin
    const int lane = threadIdx.x & 31;
    const int bin  = out % 9;
    const int no   = out / 9;
    const int n    = no >> 6, o = no & 63;

    const float* xr = xp + ((size_t)n * 9 + bin) * C_;
    const float* wr = w + (size_t)o * C_;
    float s = 0.f;
    #pragma unroll
    for (int c = 0; c < C_ / 32; ++c) s += xr[lane + 32 * c] * wr[lane + 32 * c];
    #pragma unroll
    for (int d = 16; d > 0; d >>= 1) s += __shfl_xor(s, d, 32);
    if (lane == 0) g[out] = s + b[o];
}

// ---------------------------------------------------------------------------
// Kernel 2: f = relu(conv_w(64x256) @ x[n](256xHW) + conv_b) via f32 WMMA.
// TDM stages the 256x32 x-tile into LDS once per WG (row stride 40 via D# pad),
// all 8 waves then feed B-frags from LDS (bank-conflict-free ds_load_b32).
// ---------------------------------------------------------------------------
__global__ __launch_bounds__(256) void k_conv1x1_relu(
    const float* __restrict__ x, const float* __restrict__ w,
    const float* __restrict__ b, float* __restrict__ f) {
    __shared__ float xt[C_ * LDS_STRIDE];            // 40 KB

    const int n     = blockIdx.x;
    const int ntile = blockIdx.y;                    // 0..287 (32 px each)

    // ---- wave 0: TDM load of tile x[n][0:256, col0:col0+32] -> LDS ----
    if (threadIdx.x < 32) {                          // wave-uniform branch
        const unsigned long long ga =
            (unsigned long long)(uintptr_t)(x + (size_t)n * C_ * HW_ + ntile * 32);
        const unsigned int lds = (unsigned int)(uintptr_t)(void*)xt;

        u32x4 g0;
        g0[0] = 1u;                                  // count=1 (valid D#)
        g0[1] = lds;                                 // lds_addr
        g0[2] = (unsigned int)ga;                    // global_addr[31:0]
        g0[3] = (unsigned int)((ga >> 32) & 0x01FFFFFFu) | (2u << 30); // type=2

        i32x8 g1;
        g1[0] = 0x0F120000;      // data_size=4B, pad_en, interval=32dw, amount=8dw
        g1[1] = (int)((HW_ & 0xFFFF) << 16);               // tensor_dim0 lo16
        g1[2] = (int)(((HW_ >> 16) & 0xFFFF) | ((C_ & 0xFFFF) << 16)); // d0 hi|d1 lo
        g1[3] = (int)(((C_ >> 16) & 0xFFFF) | (32 << 16)); // d1 hi | tile_dim0=32
        g1[4] = (int)(C_ & 0xFFFF);                        // tile_dim1=256
        g1[5] = (int)HW_;                                  // dim0_stride lo32
        g1[6] = 0;
        g1[7] = 0;
        i32x4 gz = {0, 0, 0, 0};
        TENSOR_LOAD_TO_LDS(g0, g1, gz, gz);
        __builtin_amdgcn_s_wait_tensorcnt(0);
    }
    __syncthreads();

    // ---- WMMA GEMM: 64(M) x 32(N), K=256 ----
    const int wave  = threadIdx.x >> 5;
    const int lane  = threadIdx.x & 31;
    const int lane_lo = lane & 15, lane_hi = lane >> 4;
    const int mbase = (wave >> 1) * 16;
    const int nb    = (wave & 1) * 16;
    const int col   = ntile * 32 + nb + lane_lo;

    const float* wrow = w + (size_t)(mbase + lane_lo) * C_;

    v8f acc = {};
    for (int k = 0; k < C_; k += 4) {
        const int kb = k + 2 * lane_hi;
        v2f a, bb;
        a.x  = wrow[kb];
        a.y  = wrow[kb + 1];
        bb.x = xt[kb * LDS_STRIDE + nb + lane_lo];
        bb.y = xt[(kb + 1) * LDS_STRIDE + nb + lane_lo];
        acc = __builtin_amdgcn_wmma_f32_16x16x4_f32(
            false, a, false, bb, (short)0, acc, false, false);
    }

    float* fout = f + (size_t)n * CM_ * HW_ + col;
    #pragma unroll
    for (int r = 0; r < 8; ++r) {
        const int m = mbase + r + 8 * lane_hi;
        float v = acc[r] + b[m];
        fout[(size_t)m * HW_] = v > 0.f ? v : 0.f;
    }
}

// ---------------------------------------------------------------------------
// Kernel 3: fused dynamic depthwise conv + fuse GEMM (f fits in 192MB L2).
// ---------------------------------------------------------------------------
__global__ __launch_bounds__(256) void k_dw_fuse(
    const float* __restrict__ f, const float* __restrict__ g,
    const float* __restrict__ dwb, const float* __restrict__ fw,
    const float* __restrict__ fb, float* __restrict__ y) {
    __shared__ float o_lds[CM_ * LDS_STRIDE];        // 10 KB
    const int ct = blockIdx.x;        // 0..2
    const int h  = blockIdx.y;        // 0..95
    const int n  = blockIdx.z;        // 0..31
    const int w0 = ct * 32;
    const int tid  = threadIdx.x;
    const int colp = tid & 31;

    // ---- Phase 1: depthwise conv (zero padding) into LDS ----
    #pragma unroll
    for (int s = 0; s < 8; ++s) {
        const int cm = (tid >> 5) + 8 * s;
        const float* fp = f + ((size_t)n * CM_ + cm) * HW_;
        const float* gp = g + ((size_t)n * CM_ + cm) * 9;
        float accv = dwb[cm];
        #pragma unroll
        for (int ky = 0; ky < 3; ++ky) {
            const int hh = h + ky - 1;
            if ((unsigned)hh < (unsigned)H_) {
                #pragma unroll
                for (int kx = 0; kx < 3; ++kx) {
                    const int ww = w0 + colp + kx - 1;
                    if ((unsigned)ww < (unsigned)W_)
                        accv += fp[hh * W_ + ww] * gp[ky * 3 + kx];
                }
            }
        }
        o_lds[cm * LDS_STRIDE + colp] = accv;
    }
    __syncthreads();

    // ---- Phase 2: 256x32 = fuse_w(256x64) @ o(64x32) via f32 WMMA ----
    const int wave = tid >> 5, lane = tid & 31;
    const int lane_lo = lane & 15, lane_hi = lane >> 4;
    #pragma unroll
    for (int t = 0; t < 4; ++t) {
        const int tile  = wave * 4 + t;       // 0..31
        const int mbase = (tile >> 1) * 16;   // 0..240
        const int nb    = (tile & 1) * 16;    // 0 or 16
        const float* arow = fw + (size_t)(mbase + lane_lo) * CM_;

        v8f acc = {};
        for (int k = 0; k < CM_; k += 4) {
            const int kb = k + 2 * lane_hi;
            v2f a, bb;
            a.x  = arow[kb];
            a.y  = arow[kb + 1];
            bb.x = o_lds[kb * LDS_STRIDE + nb + lane_lo];
            bb.y = o_lds[(kb + 1) * LDS_STRIDE + nb + lane_lo];
            acc = __builtin_amdgcn_wmma_f32_16x16x4_f32(
                false, a, false, bb, (short)0, acc, false, false);
        }

        float* yp = y + (size_t)n * P_ * HW_ + h * W_ + w0 + nb + lane_lo;
        #pragma unroll
        for (int r = 0; r < 8; ++r) {
            const int m = mbase + r + 8 * lane_hi;
            yp[(size_t)m * HW_] = acc[r] + fb[m];
        }
    }
}

// ---------------------------------------------------------------------------
extern "C" void kernel_launch(void* const* d_in, const int* in_sizes, int n_in,
                              void* d_out, int out_size, void* d_ws,
                              size_t ws_size, hipStream_t stream) {
    const float* x      = (const float*)d_in[0];
    const float* conv_w = (const float*)d_in[1];
    const float* conv_b = (const float*)d_in[2];
    const float* dw_b   = (const float*)d_in[3];
    const float* fuse_w = (const float*)d_in[4];
    const float* fuse_b = (const float*)d_in[5];
    float* y = (float*)d_out;

    // workspace: f (N*Cm*HW) | xp_t (N*9*C) | g (N*Cm*9)
    float* f_ws  = (float*)d_ws;
    float* xp_ws = f_ws + (size_t)N_ * CM_ * HW_;
    float* g_ws  = xp_ws + (size_t)N_ * 9 * C_;

    k_pool<<<N_ * C_, 288, 0, stream>>>(x, xp_ws);
    k_gfilt<<<(N_ * CM_ * 9) / 8, 256, 0, stream>>>(xp_ws, conv_w, conv_b, g_ws);

    dim3 g2(N_, HW_ / 32);
    k_conv1x1_relu<<<g2, 256, 0, stream>>>(x, conv_w, conv_b, f_ws);

    dim3 g3(3, H_, N_);
    k_dw_fuse<<<g3, 256, 0, stream>>>(f_ws, g_ws, dw_b, fuse_w, fuse_b, y);
}